// WheelSoftmax_30339648979689
// MI455X (gfx1250) — compile-verified
//
#include <hip/hip_runtime.h>
#include <hip/hip_bf16.h>

// Wheel softmax over last dim: rows of 2048 f32.
// One 256-thread block processes RPB rows; rows are streamed global->LDS with
// CDNA5 async-DMA (global_load_async_to_lds_b128) using a 2-deep double buffer
// synchronized by s_wait_asynccnt. Each thread owns 8 elements (2x float4),
// so each lane only ever reads back bytes its own async op wrote (no barrier
// needed for the copy; barriers are only used for the cross-wave reductions).

#define COLS 2048
#define TPB  256
#define RPB  16   // rows per block (pipelined)

typedef float v4f __attribute__((ext_vector_type(4)));

__device__ __forceinline__ float fin_or_zero(float v) {
    return __builtin_isfinite(v) ? v : 0.0f;
}

__global__ __launch_bounds__(TPB)
void wheel_softmax_2048(const float* __restrict__ x,
                        float* __restrict__ out,
                        int nrows) {
    // 2 x 8KB row buffers + reduction scratch
    __shared__ v4f   sbuf[2][COLS / 4];
    __shared__ float sred[2][TPB / 32];

    const int t    = threadIdx.x;
    const int wave = t >> 5;
    const int lane = t & 31;

    const int row0 = blockIdx.x * RPB;
    int rend = row0 + RPB;
    if (rend > nrows) rend = nrows;
    if (row0 >= rend) return;

    // float4 slot indices owned by this thread (two contiguous 4KB segments)
    const int s0 = t;                 // floats [4t .. 4t+3]
    const int s1 = (COLS / 8) + t;    // floats [1024+4t .. 1024+4t+3]

    auto issue_row = [&](int row, int buf) {
        const float* g0 = x + (size_t)row * COLS + (size_t)s0 * 4;
        const float* g1 = x + (size_t)row * COLS + (size_t)s1 * 4;
        unsigned l0 = (unsigned)(uintptr_t)(const void*)&sbuf[buf][s0];
        unsigned l1 = (unsigned)(uintptr_t)(const void*)&sbuf[buf][s1];
        asm volatile("global_load_async_to_lds_b128 %0, %1, off"
                     :: "v"(l0), "v"(g0) : "memory");
        asm volatile("global_load_async_to_lds_b128 %0, %1, off"
                     :: "v"(l1), "v"(g1) : "memory");
    };

    issue_row(row0, 0);

    for (int row = row0; row < rend; ++row) {
        const int buf = (row - row0) & 1;

        if (row + 1 < rend) {
            issue_row(row + 1, buf ^ 1);
            // 2 newer async ops pending; oldest 2 (this row) must be done.
            asm volatile("s_wait_asynccnt 2" ::: "memory");
        } else {
            asm volatile("s_wait_asynccnt 0" ::: "memory");
        }

        v4f a0 = sbuf[buf][s0];
        v4f a1 = sbuf[buf][s1];

        // ---- row max ----
        float m = fmaxf(fmaxf(fmaxf(a0.x, a0.y), fmaxf(a0.z, a0.w)),
                        fmaxf(fmaxf(a1.x, a1.y), fmaxf(a1.z, a1.w)));
#pragma unroll
        for (int off = 16; off > 0; off >>= 1)
            m = fmaxf(m, __shfl_xor(m, off, 32));
        if (lane == 0) sred[0][wave] = m;
        __syncthreads();
        m = sred[0][0];
#pragma unroll
        for (int w = 1; w < TPB / 32; ++w) m = fmaxf(m, sred[0][w]);

        // ---- exp + row sum ----
        v4f e0, e1;
        e0.x = __expf(a0.x - m); e0.y = __expf(a0.y - m);
        e0.z = __expf(a0.z - m); e0.w = __expf(a0.w - m);
        e1.x = __expf(a1.x - m); e1.y = __expf(a1.y - m);
        e1.z = __expf(a1.z - m); e1.w = __expf(a1.w - m);

        float s = ((e0.x + e0.y) + (e0.z + e0.w)) +
                  ((e1.x + e1.y) + (e1.z + e1.w));
#pragma unroll
        for (int off = 16; off > 0; off >>= 1)
            s += __shfl_xor(s, off, 32);
        if (lane == 0) sred[1][wave] = s;
        __syncthreads();
        s = sred[1][0];
#pragma unroll
        for (int w = 1; w < TPB / 32; ++w) s += sred[1][w];

        // ---- scale, wheel-bottom mask, streaming store ----
        const float inv = 1.0f / s;
        v4f p0, p1;
        p0.x = fin_or_zero(e0.x * inv); p0.y = fin_or_zero(e0.y * inv);
        p0.z = fin_or_zero(e0.z * inv); p0.w = fin_or_zero(e0.w * inv);
        p1.x = fin_or_zero(e1.x * inv); p1.y = fin_or_zero(e1.y * inv);
        p1.z = fin_or_zero(e1.z * inv); p1.w = fin_or_zero(e1.w * inv);

        v4f* o = (v4f*)(out + (size_t)row * COLS);
        __builtin_nontemporal_store(p0, o + s0);
        __builtin_nontemporal_store(p1, o + s1);
    }
}

extern "C" void kernel_launch(void* const* d_in, const int* in_sizes, int n_in,
                              void* d_out, int out_size, void* d_ws, size_t ws_size,
                              hipStream_t stream) {
    const float* x = (const float*)d_in[0];
    float* out     = (float*)d_out;

    const int total = in_sizes[0];          // 4*16*2048*2048
    const int nrows = total / COLS;         // 131072 rows of 2048
    const int blocks = (nrows + RPB - 1) / RPB;

    hipLaunchKernelGGL(wheel_softmax_2048, dim3(blocks), dim3(TPB), 0, stream,
                       x, out, nrows);
}